// TrilinearInterpolationModel_12601434046695
// MI455X (gfx1250) — compile-verified
//
#include <hip/hip_runtime.h>
#include <stdint.h>

// ---------------------------------------------------------------------------
// Trilinear 3D-LUT forward (image-adaptive-3DLUT) for MI455X / gfx1250.
//
// Memory-bound gather problem (no WMMA applicable: ~2 flop/byte, data-
// dependent addresses). Stage one LUT channel (143.7 KB) into LDS per block
// via CDNA5 async global->LDS copies (ASYNCcnt), then serve the 8 corner
// gathers per pixel from LDS banks instead of scattering into WGP$/L2.
// Streaming I/O is float4 (b128); blocks are persistent per channel.
// ---------------------------------------------------------------------------

#define LUT_D    33
#define LUT_D2   (LUT_D * LUT_D)            // 1089
#define LUT_D3   (LUT_D * LUT_D * LUT_D)    // 35937 floats = 143,748 B
#define HW       (1024 * 1024)
#define HW4      (HW / 4)                   // 262144 float4 per plane
#define NBATCH   4
#define NPIX     (NBATCH * HW)              // 4,194,304 pixel positions
#define NP4      (NPIX / 4)                 // 1,048,576 float4 work items / channel
#define TPB      256
#define NTILES   (NP4 / TPB)                // 4096 tiles per channel
#define GRID_X   256                        // persistent blocks per channel

// Types matching the async-copy builtin's parameters:
//   arg0: int4 in global (__device__, AS1) space, arg1: int4 in LDS (AS3).
typedef int v4i __attribute__((vector_size(16)));
typedef __attribute__((address_space(1))) v4i g_v4i;
typedef __attribute__((address_space(3))) v4i l_v4i;
typedef __attribute__((address_space(3))) void las_void;

__device__ __forceinline__ void wait_async0() {
#if __has_builtin(__builtin_amdgcn_s_wait_asynccnt)
    __builtin_amdgcn_s_wait_asynccnt(0);
#else
    asm volatile("s_wait_asynccnt 0" ::: "memory");
#endif
}

// One trilinear sample for a single output channel out of the LDS-resident
// per-channel LUT. Matches reference: s = c*32, idx = clip(floor(s),0,31),
// d = s - idx, corners at base {+0,+1,+33,+34,+1089,+1090,+1122,+1123}.
__device__ __forceinline__ float tri1(const float* __restrict__ S,
                                      float r, float g, float b) {
    float sr = r * 32.0f, sg = g * 32.0f, sb = b * 32.0f;
    float fr = fminf(fmaxf(floorf(sr), 0.0f), 31.0f);
    float fg = fminf(fmaxf(floorf(sg), 0.0f), 31.0f);
    float fb = fminf(fmaxf(floorf(sb), 0.0f), 31.0f);
    float dr = sr - fr, dg = sg - fg, db = sb - fb;
    int ir = (int)fr, ig = (int)fg, ib = (int)fb;
    int base = (ib * LUT_D + ig) * LUT_D + ir;

    float c000 = S[base];
    float c001 = S[base + 1];
    float c010 = S[base + LUT_D];
    float c011 = S[base + LUT_D + 1];
    float c100 = S[base + LUT_D2];
    float c101 = S[base + LUT_D2 + 1];
    float c110 = S[base + LUT_D2 + LUT_D];
    float c111 = S[base + LUT_D2 + LUT_D + 1];

    float a0 = fmaf(dr, c001 - c000, c000);
    float a1 = fmaf(dr, c011 - c010, c010);
    float a2 = fmaf(dr, c101 - c100, c100);
    float a3 = fmaf(dr, c111 - c110, c110);
    float b0 = fmaf(dg, a1 - a0, a0);
    float b1 = fmaf(dg, a3 - a2, a2);
    return fmaf(db, b1 - b0, b0);
}

__global__ __launch_bounds__(TPB)
void lut3d_trilerp_kernel(const float* __restrict__ lut,
                          const float* __restrict__ x,
                          float* __restrict__ out) {
    extern __shared__ float smem[];          // 35937 floats (143,748 B)
    const int tid = (int)threadIdx.x;
    const int c   = (int)blockIdx.y;         // output channel 0..2
    const float* __restrict__ lutc = lut + (size_t)c * LUT_D3;

    // ---- Stage this channel's LUT into LDS: 8984 x 16B async chunks + 1 tail
    const int NCHUNK = LUT_D3 / 4;           // 8984 (covers floats [0, 35936))
    for (int i = tid; i < NCHUNK; i += TPB) {
#if __has_builtin(__builtin_amdgcn_global_load_async_to_lds_b128)
        __builtin_amdgcn_global_load_async_to_lds_b128(
            (g_v4i*)(lutc + 4 * i),
            (l_v4i*)(smem + 4 * i),
            /*offset=*/0, /*cpol=*/0);
#else
        uint32_t loff = (uint32_t)(uintptr_t)(las_void*)(smem + 4 * i);
        const float* gp = lutc + 4 * i;
        asm volatile("global_load_async_to_lds_b128 %0, %1, off"
                     :: "v"(loff), "v"(gp) : "memory");
#endif
    }
    if (tid == 0) smem[LUT_D3 - 1] = lutc[LUT_D3 - 1];  // tail float 35936
    wait_async0();
    __syncthreads();

    // ---- Persistent loop over pixel tiles (float4 = 4 pixels per thread) ----
    const float4* __restrict__ x4 = (const float4*)x;
    float4* __restrict__ o4       = (float4*)out;
    const int stride4 = (int)gridDim.x * TPB;

    for (int t = (int)blockIdx.x; t < NTILES; t += (int)gridDim.x) {
        int p4 = t * TPB + tid;                    // [0, NP4)
        int b  = p4 / HW4;                         // batch index
        int q  = p4 - b * HW4;                     // float4 index within plane
        size_t base4 = (size_t)b * 3 * HW4;

        // Prefetch next persistent iteration's streaming reads (speculative).
        int pn = p4 + stride4;
        if (pn < NP4) {
            int bn = pn / HW4;
            int qn = pn - bn * HW4;
            size_t bb = (size_t)bn * 3 * HW4;
            __builtin_prefetch(&x4[bb + qn],           0, 0);
            __builtin_prefetch(&x4[bb + HW4 + qn],     0, 0);
            __builtin_prefetch(&x4[bb + 2 * HW4 + qn], 0, 0);
        }

        float4 rv = x4[base4 + q];                 // channel 0 plane
        float4 gv = x4[base4 + HW4 + q];           // channel 1 plane
        float4 bv = x4[base4 + 2 * HW4 + q];       // channel 2 plane

        float4 ov;
        ov.x = tri1(smem, rv.x, gv.x, bv.x);
        ov.y = tri1(smem, rv.y, gv.y, bv.y);
        ov.z = tri1(smem, rv.z, gv.z, bv.z);
        ov.w = tri1(smem, rv.w, gv.w, bv.w);

        o4[base4 + (size_t)c * HW4 + q] = ov;
    }
}

extern "C" void kernel_launch(void* const* d_in, const int* in_sizes, int n_in,
                              void* d_out, int out_size, void* d_ws, size_t ws_size,
                              hipStream_t stream) {
    (void)in_sizes; (void)n_in; (void)out_size; (void)d_ws; (void)ws_size;
    const float* lut = (const float*)d_in[0];   // [3, 33, 33, 33] f32
    const float* x   = (const float*)d_in[1];   // [4, 3, 1024, 1024] f32
    float* out = (float*)d_out;                 // [4, 3, 1024, 1024] f32

    const size_t shmem = (size_t)LUT_D3 * sizeof(float);  // 143,748 B (< 320 KB)
    // Opt in to >64KB dynamic LDS (CDNA5 WGP has 320 KB).
    (void)hipFuncSetAttribute((const void*)lut3d_trilerp_kernel,
                              hipFuncAttributeMaxDynamicSharedMemorySize,
                              (int)shmem);

    dim3 grid(GRID_X, 3, 1);
    lut3d_trilerp_kernel<<<grid, TPB, shmem, stream>>>(lut, x, out);
}